// PVConv_70153995813274
// MI455X (gfx1250) — compile-verified
//
#include <hip/hip_runtime.h>
#include <math.h>

typedef __attribute__((ext_vector_type(2))) float v2f;
typedef __attribute__((ext_vector_type(8))) float v8f;

#define R_     32
#define V_     (R_*R_*R_)     // 32768 voxels
#define B_     8
#define CIN_   64
#define COUT_  128
#define NPTS_  16384

// stats layout (floats): [0..23] coord means (b*3+d), [24..31] denom per batch,
// [64 + stage*128 + (b*8+g)*2 + {0,1}] GN sum / sumsq
#define STAT_GN_BASE 64

// ---------------------------------------------------------------- utilities
__global__ void zero_kernel(float* __restrict__ p, size_t n) {
    size_t i = (size_t)blockIdx.x * blockDim.x + threadIdx.x;
    size_t stride = (size_t)gridDim.x * blockDim.x;
    for (; i < n; i += stride) p[i] = 0.0f;
}

// -------------------------------------------------- weight repack: w[co][ci][tap] -> wr[tap][ci/2][co][2]
// Makes each WMMA A-fragment a single coalesced b64 load.
template <int CI>
__global__ __launch_bounds__(256) void repack_w_kernel(const float* __restrict__ w,
                                                       float* __restrict__ wr) {
    int p = blockIdx.x * 256 + threadIdx.x;       // over 27*(CI/2)*COUT
    int co = p % COUT_;
    int rest = p / COUT_;
    int cp = rest % (CI / 2);
    int tap = rest / (CI / 2);
    wr[(size_t)p * 2 + 0] = w[((size_t)co * CI + 2 * cp + 0) * 27 + tap];
    wr[(size_t)p * 2 + 1] = w[((size_t)co * CI + 2 * cp + 1) * 27 + tap];
}

// -------------------------------------------------- per-batch coord mean
__global__ __launch_bounds__(256) void mean_kernel(const float* __restrict__ coords,
                                                   float* __restrict__ stats) {
    int bd = blockIdx.x;          // 0..23
    int b = bd / 3, d = bd % 3;
    const float* src = coords + ((size_t)b * 3 + d) * NPTS_;
    __shared__ float red[256];
    float s = 0.0f;
    for (int i = threadIdx.x; i < NPTS_; i += 256) s += src[i];
    red[threadIdx.x] = s;
    __syncthreads();
    for (int o = 128; o > 0; o >>= 1) {
        if (threadIdx.x < o) red[threadIdx.x] += red[threadIdx.x + o];
        __syncthreads();
    }
    if (threadIdx.x == 0) stats[b * 3 + d] = red[0] * (1.0f / (float)NPTS_);
}

// -------------------------------------------------- per-batch max-norm scale
__global__ __launch_bounds__(256) void scale_kernel(const float* __restrict__ coords,
                                                    float* __restrict__ stats) {
    int b = blockIdx.x;
    float m0 = stats[b * 3 + 0], m1 = stats[b * 3 + 1], m2 = stats[b * 3 + 2];
    const float* cb = coords + (size_t)b * 3 * NPTS_;
    __shared__ float red[256];
    float mx = 0.0f;
    for (int n = threadIdx.x; n < NPTS_; n += 256) {
        float dx = cb[n] - m0;
        float dy = cb[NPTS_ + n] - m1;
        float dz = cb[2 * NPTS_ + n] - m2;
        float n2 = dx * dx + dy * dy + dz * dz;
        mx = fmaxf(mx, n2);
    }
    red[threadIdx.x] = mx;
    __syncthreads();
    for (int o = 128; o > 0; o >>= 1) {
        if (threadIdx.x < o) red[threadIdx.x] = fmaxf(red[threadIdx.x], red[threadIdx.x + o]);
        __syncthreads();
    }
    if (threadIdx.x == 0) stats[24 + b] = 2.0f * sqrtf(red[0]);
}

// -------------------------------------------------- per-point nc + voxel flat index
__global__ __launch_bounds__(256) void points_kernel(const float* __restrict__ coords,
                                                     const float* __restrict__ stats,
                                                     float* __restrict__ ncbuf,
                                                     int* __restrict__ flatbuf) {
    int p = blockIdx.x * 256 + threadIdx.x;   // 0 .. B*N-1
    int b = p >> 14, n = p & (NPTS_ - 1);
    float m0 = stats[b * 3 + 0], m1 = stats[b * 3 + 1], m2 = stats[b * 3 + 2];
    float inv_denom = 1.0f / stats[24 + b];
    const float* cb = coords + (size_t)b * 3 * NPTS_;
    float c0 = (cb[n]             - m0) * inv_denom + 0.5f;
    float c1 = (cb[NPTS_ + n]     - m1) * inv_denom + 0.5f;
    float c2 = (cb[2 * NPTS_ + n] - m2) * inv_denom + 0.5f;
    c0 = fminf(fmaxf(c0 * (float)R_, 0.0f), (float)(R_ - 1));
    c1 = fminf(fmaxf(c1 * (float)R_, 0.0f), (float)(R_ - 1));
    c2 = fminf(fmaxf(c2 * (float)R_, 0.0f), (float)(R_ - 1));
    int i0 = min(max((int)rintf(c0), 0), R_ - 1);
    int i1 = min(max((int)rintf(c1), 0), R_ - 1);
    int i2 = min(max((int)rintf(c2), 0), R_ - 1);
    ncbuf[(size_t)p * 3 + 0] = c0;
    ncbuf[(size_t)p * 3 + 1] = c1;
    ncbuf[(size_t)p * 3 + 2] = c2;
    flatbuf[p] = (i0 * R_ + i1) * R_ + i2;
}

// -------------------------------------------------- scatter-add features into voxels
// sums layout: channel-last [b][voxel][ci]  (coalesced feature reads, atomic scatter)
__global__ __launch_bounds__(256) void scatter_kernel(const float* __restrict__ feat,
                                                      const int* __restrict__ flatbuf,
                                                      float* __restrict__ sums,
                                                      float* __restrict__ cnts) {
    int p = blockIdx.x * 256 + threadIdx.x;
    int b = p >> 14, n = p & (NPTS_ - 1);
    int f = flatbuf[p];
    float* dst = sums + ((size_t)b * V_ + f) * CIN_;
    const float* src = feat + (size_t)b * CIN_ * NPTS_ + n;
    #pragma unroll 8
    for (int ci = 0; ci < CIN_; ++ci)
        atomicAdd(&dst[ci], src[(size_t)ci * NPTS_]);
    atomicAdd(&cnts[(size_t)b * V_ + f], 1.0f);
}

// -------------------------------------------------- average + transpose to pair-interleaved
// vox0 layout: [b][ci/2][v][2]  (WMMA B fragments become single coalesced b64 loads)
__global__ __launch_bounds__(256) void avg_kernel(const float* __restrict__ sums,
                                                  const float* __restrict__ cnts,
                                                  float* __restrict__ vox0) {
    int p = blockIdx.x * 256 + threadIdx.x;   // 0 .. B*V-1
    int b = p >> 15, v = p & (V_ - 1);
    float inv = 1.0f / fmaxf(cnts[p], 1.0f);
    const v2f* s2 = (const v2f*)(sums + (size_t)p * CIN_);
    v2f* d2 = (v2f*)vox0;
    size_t base = (size_t)b * (CIN_ / 2) * V_ + v;
    #pragma unroll 8
    for (int cp = 0; cp < CIN_ / 2; ++cp) {
        v2f val = s2[cp];
        val.x *= inv;
        val.y *= inv;
        d2[base + (size_t)cp * V_] = val;
    }
}

// -------------------------------------------------- 3x3x3 conv via V_WMMA_F32_16X16X4_F32
// One wave: 16-voxel z-strip x 128 output channels (8 co-tiles).
// A (16x4): lane l -> M=l%16, K=(l/16)*2+vgpr.  B (4x16): lane l -> N=l%16, K=(l/16)*2+vgpr.
// Input grid pair-interleaved [b][cp][v][2]; weights repacked [tap][cp][co][2].
template <int CI>
__global__ __launch_bounds__(256) void conv3_wmma(const float* __restrict__ vin,
                                                  const float* __restrict__ wr,
                                                  const float* __restrict__ bias,
                                                  float* __restrict__ out,
                                                  float* __restrict__ gstats) {
    int wv   = blockIdx.x * 8 + (threadIdx.x >> 5);   // 16384 waves
    int lane = threadIdx.x & 31;
    int b = wv >> 11;                 // 2048 tiles per batch
    int t = wv & 2047;
    int z0 = (t & 1) * 16;
    int y  = (t >> 1) & 31;
    int x  = t >> 6;
    int n  = lane & 15;
    int kg = lane >> 4;

    v8f acc[8];
    const v8f vz8 = {0.f, 0.f, 0.f, 0.f, 0.f, 0.f, 0.f, 0.f};
    #pragma unroll
    for (int i = 0; i < 8; ++i) acc[i] = vz8;
    const v2f vz2 = {0.f, 0.f};

    const v2f* vb2 = (const v2f*)vin + (size_t)b * (CI / 2) * V_;
    const v2f* wr2 = (const v2f*)wr;

    for (int kd = 0; kd < 3; ++kd) {
        int xx = x + kd - 1;
        if ((unsigned)xx >= (unsigned)R_) continue;        // wave-uniform skip
        for (int kh = 0; kh < 3; ++kh) {
            int yy = y + kh - 1;
            if ((unsigned)yy >= (unsigned)R_) continue;    // wave-uniform skip
            int basev = (xx * R_ + yy) * R_;
            for (int kw = 0; kw < 3; ++kw) {
                int zz = z0 + n + kw - 1;                  // per-lane
                bool zok = (unsigned)zz < (unsigned)R_;
                int zs = zok ? zz : 0;
                int tap = (kd * 3 + kh) * 3 + kw;
                #pragma unroll 4
                for (int cik = 0; cik < CI / 4; ++cik) {
                    int cp = cik * 2 + kg;                 // ci pair index (K pair)
                    v2f bf = zok ? vb2[(size_t)cp * V_ + basev + zs] : vz2;
                    const v2f* wt = wr2 + ((size_t)tap * (CI / 2) + cp) * COUT_;
                    #pragma unroll
                    for (int cot = 0; cot < 8; ++cot) {
                        int co = cot * 16 + n;             // M = lane%16
                        v2f af = wt[co];
                        acc[cot] = __builtin_amdgcn_wmma_f32_16x16x4_f32(
                            false, af, false, bf, (short)0, acc[cot], false, false);
                    }
                }
            }
        }
    }

    // epilogue: bias, store raw conv output (planar), fused GN partial sums (co-tile == group)
    int v0 = (x * R_ + y) * R_ + z0;
    #pragma unroll
    for (int cot = 0; cot < 8; ++cot) {
        float s = 0.0f, q = 0.0f;
        #pragma unroll
        for (int j = 0; j < 8; ++j) {
            int m = (lane < 16) ? j : (8 + j);             // D row
            int co = cot * 16 + m;
            float val = acc[cot][j] + bias[co];
            out[((size_t)b * COUT_ + co) * V_ + v0 + n] = val;
            s += val;
            q += val * val;
        }
        for (int off = 16; off >= 1; off >>= 1) {
            s += __shfl_xor(s, off, 32);
            q += __shfl_xor(q, off, 32);
        }
        if (lane == 0) {
            atomicAdd(&gstats[(b * 8 + cot) * 2 + 0], s);
            atomicAdd(&gstats[(b * 8 + cot) * 2 + 1], q);
        }
    }
}

// -------------------------------------------------- GroupNorm + swish, planar in -> pair-interleaved out
__global__ __launch_bounds__(256) void gn_swish_pack_kernel(const float* __restrict__ in,
                                                            float* __restrict__ outp,
                                                            const float* __restrict__ gamma,
                                                            const float* __restrict__ beta,
                                                            const float* __restrict__ gstats,
                                                            float invcnt) {
    size_t p = (size_t)blockIdx.x * 256 + threadIdx.x;    // B * (COUT/2) * V
    int b = (int)(p >> 21);                               // (COUT/2)*V = 2^21
    size_t r = p & ((1u << 21) - 1);
    int cp = (int)(r >> 15);                              // V_ = 2^15
    size_t v = r & (V_ - 1);
    int co0 = 2 * cp;
    int g = co0 >> 4;                                     // pair never crosses a group
    float s = gstats[(b * 8 + g) * 2 + 0];
    float q = gstats[(b * 8 + g) * 2 + 1];
    float mean = s * invcnt;
    float var  = q * invcnt - mean * mean;
    float istd = rsqrtf(var + 1e-5f);
    float x0 = in[((size_t)b * COUT_ + co0) * V_ + v];
    float x1 = in[((size_t)b * COUT_ + co0 + 1) * V_ + v];
    float y0 = (x0 - mean) * istd * gamma[co0] + beta[co0];
    float y1 = (x1 - mean) * istd * gamma[co0 + 1] + beta[co0 + 1];
    v2f o;
    o.x = y0 / (1.0f + expf(-y0));
    o.y = y1 / (1.0f + expf(-y1));
    ((v2f*)outp)[p] = o;
}

// -------------------------------------------------- point branch GEMM (wp @ features) via WMMA
__global__ __launch_bounds__(256) void pf_wmma(const float* __restrict__ feat,
                                               const float* __restrict__ wp,
                                               const float* __restrict__ bp,
                                               float* __restrict__ out,
                                               float* __restrict__ gstats) {
    int wv   = blockIdx.x * 8 + (threadIdx.x >> 5);       // 8192 waves
    int lane = threadIdx.x & 31;
    int b = wv >> 10;                                     // 1024 tiles per batch
    int t = wv & 1023;
    int n0 = t * 16;
    int n  = lane & 15;
    int kg = lane >> 4;

    v8f acc[8];
    const v8f vz8 = {0.f, 0.f, 0.f, 0.f, 0.f, 0.f, 0.f, 0.f};
    #pragma unroll
    for (int i = 0; i < 8; ++i) acc[i] = vz8;

    const float* fb = feat + (size_t)b * CIN_ * NPTS_;
    const v2f* wp2 = (const v2f*)wp;                      // wp[co][ci]: ci pairs contiguous
    #pragma unroll 4
    for (int cik = 0; cik < CIN_ / 4; ++cik) {
        int cib = cik * 4 + kg * 2;
        v2f bf;
        bf.x = fb[(size_t)(cib + 0) * NPTS_ + n0 + n];
        bf.y = fb[(size_t)(cib + 1) * NPTS_ + n0 + n];
        #pragma unroll
        for (int cot = 0; cot < 8; ++cot) {
            int co = cot * 16 + n;
            v2f af = wp2[(size_t)co * (CIN_ / 2) + (cib >> 1)];
            acc[cot] = __builtin_amdgcn_wmma_f32_16x16x4_f32(
                false, af, false, bf, (short)0, acc[cot], false, false);
        }
    }

    #pragma unroll
    for (int cot = 0; cot < 8; ++cot) {
        float s = 0.0f, q = 0.0f;
        #pragma unroll
        for (int j = 0; j < 8; ++j) {
            int m = (lane < 16) ? j : (8 + j);
            int co = cot * 16 + m;
            float val = acc[cot][j] + bp[co];
            out[((size_t)b * COUT_ + co) * NPTS_ + n0 + n] = val;
            s += val;
            q += val * val;
        }
        for (int off = 16; off >= 1; off >>= 1) {
            s += __shfl_xor(s, off, 32);
            q += __shfl_xor(q, off, 32);
        }
        if (lane == 0) {
            atomicAdd(&gstats[(b * 8 + cot) * 2 + 0], s);
            atomicAdd(&gstats[(b * 8 + cot) * 2 + 1], q);
        }
    }
}

// -------------------------------------------------- trilinear devox + point-branch GN/swish + add
// vox is pair-interleaved [b][cp][v][2] -> gathers become b64 (half the loads)
__global__ __launch_bounds__(256) void final_kernel(const float* __restrict__ vox,
                                                    const float* __restrict__ pfraw,
                                                    const float* __restrict__ ncbuf,
                                                    const float* __restrict__ gstats2,
                                                    const float* __restrict__ gpg,
                                                    const float* __restrict__ gpb,
                                                    float* __restrict__ out) {
    int p = blockIdx.x * 256 + threadIdx.x;   // B*N
    int b = p >> 14, n = p & (NPTS_ - 1);

    float cx = fminf(ncbuf[(size_t)p * 3 + 0], (float)(R_ - 1) - 1e-6f);
    float cy = fminf(ncbuf[(size_t)p * 3 + 1], (float)(R_ - 1) - 1e-6f);
    float cz = fminf(ncbuf[(size_t)p * 3 + 2], (float)(R_ - 1) - 1e-6f);
    float fx = floorf(cx), fy = floorf(cy), fz = floorf(cz);
    int ix0 = (int)fx, iy0 = (int)fy, iz0 = (int)fz;
    float dx = cx - fx, dy = cy - fy, dz = cz - fz;
    int ix1 = min(ix0 + 1, R_ - 1), iy1 = min(iy0 + 1, R_ - 1), iz1 = min(iz0 + 1, R_ - 1);

    int c000 = (ix0 * R_ + iy0) * R_ + iz0;
    int c001 = (ix0 * R_ + iy0) * R_ + iz1;
    int c010 = (ix0 * R_ + iy1) * R_ + iz0;
    int c011 = (ix0 * R_ + iy1) * R_ + iz1;
    int c100 = (ix1 * R_ + iy0) * R_ + iz0;
    int c101 = (ix1 * R_ + iy0) * R_ + iz1;
    int c110 = (ix1 * R_ + iy1) * R_ + iz0;
    int c111 = (ix1 * R_ + iy1) * R_ + iz1;
    float w000 = (1 - dx) * (1 - dy) * (1 - dz);
    float w001 = (1 - dx) * (1 - dy) * dz;
    float w010 = (1 - dx) * dy * (1 - dz);
    float w011 = (1 - dx) * dy * dz;
    float w100 = dx * (1 - dy) * (1 - dz);
    float w101 = dx * (1 - dy) * dz;
    float w110 = dx * dy * (1 - dz);
    float w111 = dx * dy * dz;

    const float invcnt = 1.0f / (16.0f * (float)NPTS_);
    float mean[8], istd[8];
    #pragma unroll
    for (int g = 0; g < 8; ++g) {
        float s = gstats2[(b * 8 + g) * 2 + 0];
        float q = gstats2[(b * 8 + g) * 2 + 1];
        float m = s * invcnt;
        mean[g] = m;
        istd[g] = rsqrtf(q * invcnt - m * m + 1e-5f);
    }

    const v2f* vb2 = (const v2f*)vox + (size_t)b * (COUT_ / 2) * V_;
    const float* pb = pfraw + (size_t)b * COUT_ * NPTS_ + n;
    float* ob = out + (size_t)b * COUT_ * NPTS_ + n;

    for (int cp = 0; cp < COUT_ / 2; ++cp) {
        const v2f* vc = vb2 + (size_t)cp * V_;
        v2f dv = w000 * vc[c000] + w001 * vc[c001] + w010 * vc[c010] + w011 * vc[c011] +
                 w100 * vc[c100] + w101 * vc[c101] + w110 * vc[c110] + w111 * vc[c111];
        int co0 = 2 * cp;
        int g = cp >> 3;
        float x0 = pb[(size_t)co0 * NPTS_];
        float x1 = pb[(size_t)(co0 + 1) * NPTS_];
        float y0 = (x0 - mean[g]) * istd[g] * gpg[co0] + gpb[co0];
        float y1 = (x1 - mean[g]) * istd[g] * gpg[co0 + 1] + gpb[co0 + 1];
        float pf0 = y0 / (1.0f + expf(-y0));
        float pf1 = y1 / (1.0f + expf(-y1));
        ob[(size_t)co0 * NPTS_]       = dv.x + pf0;
        ob[(size_t)(co0 + 1) * NPTS_] = dv.y + pf1;
    }
}

// ================================================================ host launcher
extern "C" void kernel_launch(void* const* d_in, const int* in_sizes, int n_in,
                              void* d_out, int out_size, void* d_ws, size_t ws_size,
                              hipStream_t stream) {
    const float* features = (const float*)d_in[0];
    const float* coords   = (const float*)d_in[1];
    // d_in[2] = temb (unused)
    const float* w1  = (const float*)d_in[3];
    const float* b1  = (const float*)d_in[4];
    const float* g1g = (const float*)d_in[5];
    const float* g1b = (const float*)d_in[6];
    const float* w2  = (const float*)d_in[7];
    const float* b2  = (const float*)d_in[8];
    const float* g2g = (const float*)d_in[9];
    const float* g2b = (const float*)d_in[10];
    const float* wp  = (const float*)d_in[11];
    const float* bp  = (const float*)d_in[12];
    const float* gpg = (const float*)d_in[13];
    const float* gpb = (const float*)d_in[14];
    float* out = (float*)d_out;

    // workspace partition (float offsets; all even -> 8B-aligned v2f buffers)
    float* ws = (float*)d_ws;
    size_t o = 0;
    float* stats = ws + o;              o += 1024;
    float* ncbuf = ws + o;              o += (size_t)B_ * NPTS_ * 3;
    int*   flatb = (int*)(ws + o);      o += (size_t)B_ * NPTS_;
    float* cnts  = ws + o;              o += (size_t)B_ * V_;
    float* bufA  = ws + o;              o += (size_t)B_ * V_ * CIN_;   // sums, later pf raw (same count)
    float* vox0  = ws + o;              o += (size_t)B_ * CIN_ * V_;   // pair-interleaved conv1 input
    float* bufB  = ws + o;              o += (size_t)B_ * COUT_ * V_;  // raw conv out (planar)
    float* bufC  = ws + o;              o += (size_t)B_ * COUT_ * V_;  // normalized grid (pair-interleaved)
    float* wr1   = ws + o;              o += (size_t)COUT_ * CIN_ * 27;
    float* wr2   = ws + o;              o += (size_t)COUT_ * COUT_ * 27;

    float* gn0 = stats + STAT_GN_BASE;        // conv1 GN sums
    float* gn1 = stats + STAT_GN_BASE + 128;  // conv2 GN sums
    float* gn2 = stats + STAT_GN_BASE + 256;  // pf GN sums

    // zero stats + counts + voxel sums (contiguous from ws start through bufA)
    size_t nzero = 1024 + (size_t)B_ * NPTS_ * 3 + (size_t)B_ * NPTS_ +
                   (size_t)B_ * V_ + (size_t)B_ * V_ * CIN_;
    zero_kernel<<<4096, 256, 0, stream>>>(ws, nzero);

    // weight repacks (once per launch; tiny)
    repack_w_kernel<CIN_><<<(27 * (CIN_ / 2) * COUT_) / 256, 256, 0, stream>>>(w1, wr1);
    repack_w_kernel<COUT_><<<(27 * (COUT_ / 2) * COUT_) / 256, 256, 0, stream>>>(w2, wr2);

    mean_kernel<<<24, 256, 0, stream>>>(coords, stats);
    scale_kernel<<<8, 256, 0, stream>>>(coords, stats);
    points_kernel<<<(B_ * NPTS_) / 256, 256, 0, stream>>>(coords, stats, ncbuf, flatb);
    scatter_kernel<<<(B_ * NPTS_) / 256, 256, 0, stream>>>(features, flatb, bufA, cnts);
    avg_kernel<<<(B_ * V_) / 256, 256, 0, stream>>>(bufA, cnts, vox0);

    // conv1 (64 -> 128) + fused GN partial sums
    conv3_wmma<CIN_><<<(B_ * (V_ / 16)) / 8, 256, 0, stream>>>(vox0, wr1, b1, bufB, gn0);
    gn_swish_pack_kernel<<<(int)(((size_t)B_ * (COUT_ / 2) * V_) / 256), 256, 0, stream>>>(
        bufB, bufC, g1g, g1b, gn0, 1.0f / (16.0f * (float)V_));

    // conv2 (128 -> 128)
    conv3_wmma<COUT_><<<(B_ * (V_ / 16)) / 8, 256, 0, stream>>>(bufC, wr2, b2, bufB, gn1);
    gn_swish_pack_kernel<<<(int)(((size_t)B_ * (COUT_ / 2) * V_) / 256), 256, 0, stream>>>(
        bufB, bufC, g2g, g2b, gn1, 1.0f / (16.0f * (float)V_));
    // NOTE: bufC now holds the normalized conv2 grid, pair-interleaved — exactly what final_kernel reads.

    // point branch GEMM (reuses bufA — voxel sums no longer needed)
    pf_wmma<<<(B_ * (NPTS_ / 16)) / 8, 256, 0, stream>>>(features, wp, bp, bufA, gn2);

    // devoxelize + point GN/swish + fuse
    final_kernel<<<(B_ * NPTS_) / 256, 256, 0, stream>>>(bufC, bufA, ncbuf, gn2, gpg, gpb, out);
}